// Nullary_49950469653356
// MI455X (gfx1250) — compile-verified
//
#include <hip/hip_runtime.h>

typedef __attribute__((ext_vector_type(2))) float v2f;
typedef __attribute__((ext_vector_type(8))) float v8f;

#define D 64
#define NW_TILE 64
#define LDS_STRIDE 68   // 272B rows: 16B-aligned, conflict-free b64 fragment reads

__global__ void zero_f4(float4* __restrict__ p, long long n4) {
  long long i = (long long)blockIdx.x * blockDim.x + threadIdx.x;
  if (i < n4) p[i] = make_float4(0.f, 0.f, 0.f, 0.f);
}

__global__ __launch_bounds__(512)
void nullary_norm_scatter(const float* __restrict__ worlds,
                          const float* __restrict__ W,
                          const int*   __restrict__ nullary,
                          float*       __restrict__ out,
                          int n_worlds) {
  __shared__ __align__(16) float As[D * LDS_STRIDE];        // W[bsel]: [d][e]
  __shared__ __align__(16) float Bs[NW_TILE * LDS_STRIDE];  // worlds tile: [w][e]
  __shared__ float sqp[4][NW_TILE];                          // per-mi column partials

  const int tid  = threadIdx.x;
  const int lane = tid & 31;
  const int wave = tid >> 5;
  const int lo   = lane & 15;
  const int hi   = lane >> 4;
  const int mi   = wave >> 2;   // 0..3 : d-block
  const int ni   = wave & 3;    // 0..3 : w-block

  const int n     = blockIdx.y;
  const int wbase = blockIdx.x * NW_TILE;
  const int bsel  = nullary[2 * n + 1];
  const int asel  = nullary[2 * n + 0];

  // Stage W[bsel] (64x64) and the worlds tile (64x64) into LDS, float4 wide.
  const float4* Wsrc = (const float4*)(W + (size_t)bsel * D * D);
  const float4* Usrc = (const float4*)(worlds + (size_t)wbase * D);
  for (int f = tid; f < (D * D) / 4; f += 512) {
    int row = f >> 4;
    int col = (f & 15) << 2;
    *(float4*)&As[row * LDS_STRIDE + col] = Wsrc[f];
    *(float4*)&Bs[row * LDS_STRIDE + col] = Usrc[f];
  }
  __syncthreads();

  // Per-lane fragment rows (ISA 7.12.2 layouts for 16x4 A / 4x16 B f32).
  const float* Arow = &As[(mi * 16 + lo) * LDS_STRIDE];
  const float* Brow = &Bs[(ni * 16 + lo) * LDS_STRIDE];

  v8f c = {};
#pragma unroll
  for (int k = 0; k < D; k += 4) {
    const int kk = k + 2 * hi;                 // even offset -> 8B aligned
    v2f a = *(const v2f*)(Arow + kk);          // A[m][kk], A[m][kk+1]
    v2f b = *(const v2f*)(Brow + kk);          // B[kk][nw], B[kk+1][nw]
    c = __builtin_amdgcn_wmma_f32_16x16x4_f32(false, a, false, b,
                                              (short)0, c, false, false);
  }

  // Column sum-of-squares: fold 8 rows in-lane, then the two row-halves
  // across the wave32 (xor-16 shuffle), then the 4 mi blocks via LDS.
  float s = 0.f;
#pragma unroll
  for (int r = 0; r < 8; ++r) s += c[r] * c[r];
  s += __shfl_xor(s, 16, 32);
  if (lane < 16) sqp[mi][ni * 16 + lo] = s;
  __syncthreads();

  const int col   = ni * 16 + lo;
  const float tot = sqp[0][col] + sqp[1][col] + sqp[2][col] + sqp[3][col];
  const float scale = rsqrtf(fmaxf(tot, 1e-12f));

  // Scatter-add the normalized tile: out[asel][m][wbase+col]
  float* outp = out + (((size_t)asel * D) + (size_t)(mi * 16 + 8 * hi)) * n_worlds
                    + (wbase + col);
#pragma unroll
  for (int r = 0; r < 8; ++r)
    atomicAdd(outp + (size_t)r * n_worlds, c[r] * scale);
}

extern "C" void kernel_launch(void* const* d_in, const int* in_sizes, int n_in,
                              void* d_out, int out_size, void* d_ws, size_t ws_size,
                              hipStream_t stream) {
  const float* worlds  = (const float*)d_in[0];   // (2048, 64) f32
  const float* W       = (const float*)d_in[1];   // (512, 64, 64) f32
  const int*   nullary = (const int*)d_in[2];     // (1024, 2) i32
  float* out = (float*)d_out;                     // (1024, 64, 2048) f32

  const int n_worlds = in_sizes[0] / D;           // 2048
  const int batch    = in_sizes[2] / 2;           // 1024

  // Zero the (poisoned) output before scatter-add.
  long long n4 = (long long)out_size / 4;
  int zblocks = (int)((n4 + 255) / 256);
  zero_f4<<<zblocks, 256, 0, stream>>>((float4*)d_out, n4);

  dim3 grid(n_worlds / NW_TILE, batch);           // (32, 1024)
  nullary_norm_scatter<<<grid, 512, 0, stream>>>(worlds, W, nullary, out, n_worlds);
}